// AnisotropicStack_23716809408986
// MI455X (gfx1250) — compile-verified
//
#include <hip/hip_runtime.h>
#include <hip/hip_bf16.h>

// Problem constants (match reference)
#define B_  8
#define L_  8192
#define D_  1024
#define M_  (L_ / 4)        // 2048 selected tokens per batch row (enforced by setup)
#define TSTEPS 16           // scan-tile length -> 16x16 triangular matmul via WMMA
#define NT  (M_ / TSTEPS)   // 128 tiles per batch row

typedef float v2f __attribute__((ext_vector_type(2)));
typedef float v8f __attribute__((ext_vector_type(8)));

// ---------------------------------------------------------------------------
// Kernel 1: tokenizer. One block per batch row. Block-wide prefix sum of the
// mask produces, for each selected position (stable order):
//   sel_idx[b][j]   = source position l of j-th selected token (+ sentinel L)
//   sel_decay[b][j] = clamp(1 - prob[b][l], 0, 1)
// ---------------------------------------------------------------------------
__global__ __launch_bounds__(512)
void ema_tokenize_kernel(const int* __restrict__ mask,
                         const float* __restrict__ prob,
                         int* __restrict__ sel_idx,
                         float* __restrict__ sel_decay) {
    const int b   = blockIdx.x;
    const int tid = threadIdx.x;            // 0..511, each owns 16 positions
    const int base = tid * 16;

    __shared__ int scn[512];

    const int* mrow = mask + (size_t)b * L_;
    int m16 = 0, cnt = 0;
#pragma unroll
    for (int i = 0; i < 16; ++i) {
        int mm = (mrow[base + i] != 0);
        m16 |= mm << i;
        cnt += mm;
    }
    scn[tid] = cnt;
    __syncthreads();
    // Hillis-Steele inclusive scan over the 512 per-thread counts
    for (int off = 1; off < 512; off <<= 1) {
        int v   = scn[tid];
        int add = (tid >= off) ? scn[tid - off] : 0;
        __syncthreads();
        scn[tid] = v + add;
        __syncthreads();
    }
    int j = scn[tid] - cnt;                 // exclusive prefix
    const float* prow = prob + (size_t)b * L_;
#pragma unroll
    for (int i = 0; i < 16; ++i) {
        if ((m16 >> i) & 1) {
            const int l = base + i;
            sel_idx[(size_t)b * (M_ + 1) + j] = l;
            float dcy = 1.0f - prow[l];
            dcy = fminf(fmaxf(dcy, 0.0f), 1.0f);
            sel_decay[(size_t)b * M_ + j] = dcy;
            ++j;
        }
    }
    if (tid == 0) sel_idx[(size_t)b * (M_ + 1) + M_] = L_;  // sentinel
}

// ---------------------------------------------------------------------------
// Kernel 2: fused gather + chunked EMA scan (WMMA) + scatter-expand.
// Block = 128 threads (4 wave32), owns (batch b, 64 feature columns).
// Per 16-step tile:
//   T[j][k] = prod_{i=k+1..j} a_i (lower-triangular), cp[j] = prod_{i<=j} a_i
//   S (16x64) = T (16x16) x Bv (16x64)  via 4x v_wmma_f32_16x16x4_f32 per wave
//   H[j][d]  = S[j][d] + cp[j] * h_entry[d]
// then all output rows covered by these 16 chunks are written:
//   out[l] = residual[l] + H[chunk(l)]    (fused, no ema tensor in HBM)
// ---------------------------------------------------------------------------
__global__ __launch_bounds__(128)
void ema_scan_scatter_kernel(const float* __restrict__ x,
                             const float* __restrict__ residual,
                             const float* __restrict__ state,
                             const int* __restrict__ sel_idx,
                             const float* __restrict__ sel_decay,
                             float* __restrict__ out,         // (B, L, D)
                             float* __restrict__ new_state) { // (B, D)
    const int b    = blockIdx.y;
    const int d0   = blockIdx.x * 64;       // block's feature window
    const int tid  = threadIdx.x;           // 0..127
    const int wave = tid >> 5;              // 0..3 (wave32)
    const int lane = tid & 31;
    const int half = lane >> 4;             // 0: lanes 0-15, 1: lanes 16-31
    const int ln   = lane & 15;
    const int dw   = d0 + wave * 16 + ln;   // lane's feature column (replicated halves)

    __shared__ float Ts[16][16];            // triangular transition matrix
    __shared__ float aS[16];                // decay coefficients of this tile
    __shared__ float cpS[16];               // cumulative products
    __shared__ int   selS[17];              // chunk -> source-row boundaries
    __shared__ __align__(16) float Ht[16][64];  // H tile staged for scatter

    const float* xb  = x        + (size_t)b * L_ * D_;
    const float* rb  = residual + (size_t)b * L_ * D_;
    float*       ob  = out      + (size_t)b * L_ * D_;
    const int*   sib = sel_idx  + (size_t)b * (M_ + 1);
    const float* sdb = sel_decay + (size_t)b * M_;

    float hprev = state[(size_t)b * D_ + dw];   // h_{-1} = state

    // Pre-span: positions before the first selected token -> out = residual
    {
        const int end = sib[0];
        const int g16 = tid >> 4, l16 = tid & 15;   // 8 rows x 16 float4 lanes
        for (int l = g16; l < end; l += 8) {
            const float4* r4 = (const float4*)(rb + (size_t)l * D_ + d0);
            float4*       o4 = (float4*)(ob + (size_t)l * D_ + d0);
            o4[l16] = r4[l16];
        }
    }

    for (int t = 0; t < NT; ++t) {
        const int j0 = t * TSTEPS;

        // ---- phase 1a: fetch tile coefficients + chunk boundaries ----
        if (tid < 16) aS[tid]  = sdb[j0 + tid];
        if (tid < 17) selS[tid] = sib[j0 + tid];
        __syncthreads();

        // ---- phase 1b: build T and cp (16 threads, one row each) ----
        if (tid < 16) {
            const int j = tid;
            float prod = 1.0f;
            Ts[j][j] = 1.0f;
            for (int k = j - 1; k >= 0; --k) {
                prod *= aS[k + 1];
                Ts[j][k] = prod;            // prod_{i=k+1..j} a_i
            }
            for (int k = j + 1; k < 16; ++k) Ts[j][k] = 0.0f;
            cpS[j] = prod * aS[0];          // prod_{i=0..j} a_i
        }
        __syncthreads();

        // ---- phase 2: S = T x Bv on the matrix pipe (EXEC all-ones) ----
        // A (16x4 f32) lane layout: lanes0-15 hold K0(K1 in v1), lanes16-31 K2/K3.
        // B (4x16 f32) mirrored:    lanes0-15 hold rows K0/K1,   lanes16-31 K2/K3.
        v8f c = {};
#pragma unroll
        for (int s = 0; s < 4; ++s) {
            const int k0 = 4 * s + 2 * half;
            v2f af, bf;
            af.x = Ts[ln][k0];
            af.y = Ts[ln][k0 + 1];
            const int p0 = selS[k0], p1 = selS[k0 + 1];
            bf.x = (1.0f - aS[k0])     * xb[(size_t)p0 * D_ + dw];
            bf.y = (1.0f - aS[k0 + 1]) * xb[(size_t)p1 * D_ + dw];
            c = __builtin_amdgcn_wmma_f32_16x16x4_f32(
                    false, af, false, bf, (short)0, c, false, false);
        }
        // C/D layout: lanes0-15 VGPR r -> row r; lanes16-31 VGPR r -> row r+8.
#pragma unroll
        for (int r = 0; r < 8; ++r) {
            const int j = r + 8 * half;
            Ht[j][wave * 16 + ln] = c[r] + cpS[j] * hprev;
        }
        __syncthreads();
        hprev = Ht[15][wave * 16 + ln];     // carry h_entry to next tile

        // ---- phase 3: scatter-expand the 16 chunks of this tile ----
        {
            const int g16 = tid >> 4, l16 = tid & 15;
#pragma unroll 1
            for (int j = 0; j < 16; ++j) {
                const int st = selS[j], en = selS[j + 1];
                const float4 h4 = ((const float4*)&Ht[j][0])[l16];
                for (int l = st + g16; l < en; l += 8) {
                    const float4* r4 = (const float4*)(rb + (size_t)l * D_ + d0);
                    float4*       o4 = (float4*)(ob + (size_t)l * D_ + d0);
                    float4 rv = r4[l16];
                    float4 ov = { rv.x + h4.x, rv.y + h4.y, rv.z + h4.z, rv.w + h4.w };
                    o4[l16] = ov;
                }
            }
        }
        __syncthreads();                    // protect Ts/selS/Ht reuse
    }

    // new_state = ema[:, M-1, :] (counts == M guaranteed by setup)
    if (half == 0) new_state[(size_t)b * D_ + dw] = hprev;
}

// ---------------------------------------------------------------------------
extern "C" void kernel_launch(void* const* d_in, const int* in_sizes, int n_in,
                              void* d_out, int out_size, void* d_ws, size_t ws_size,
                              hipStream_t stream) {
    (void)in_sizes; (void)n_in; (void)out_size; (void)ws_size;

    const float* x        = (const float*)d_in[0];   // (B, L, D) f32
    const float* residual = (const float*)d_in[1];   // (B, L, D) f32
    const float* prob     = (const float*)d_in[2];   // (B, L)    f32
    const int*   mask     = (const int*)  d_in[3];   // (B, L)    bool -> int
    const float* state    = (const float*)d_in[4];   // (B, D)    f32

    float* out       = (float*)d_out;                      // outputs concatenated:
    float* new_state = out + (size_t)B_ * L_ * D_;         // (B,L,D) then (B,D)

    // workspace: sel_idx (B x (M+1) ints) then sel_decay (B x M floats)
    int*   sel_idx   = (int*)d_ws;
    size_t off       = ((size_t)B_ * (M_ + 1) * sizeof(int) + 255u) & ~(size_t)255u;
    float* sel_decay = (float*)((char*)d_ws + off);

    hipLaunchKernelGGL(ema_tokenize_kernel, dim3(B_), dim3(512), 0, stream,
                       mask, prob, sel_idx, sel_decay);

    hipLaunchKernelGGL(ema_scan_scatter_kernel, dim3(D_ / 64, B_), dim3(128), 0, stream,
                       x, residual, state, sel_idx, sel_decay, out, new_state);
}